// RotaryPosEmbedReal_63290638074275
// MI455X (gfx1250) — compile-verified
//
#include <hip/hip_runtime.h>

typedef float v4f __attribute__((ext_vector_type(4)));
typedef int   v4i __attribute__((ext_vector_type(4)));

#define DEV __device__ __forceinline__

constexpr int B    = 8;
constexpr int D    = 128;    // 32 + 48 + 48
constexpr int ENC  = 512;
constexpr int REFT = 32768;  // 2 ref images, each (256/2)^2 = 16384 tokens
constexpr int IMGT = 65536;  // (512/2)^2
constexpr int D0 = 32, D1 = 48, D2 = 48;

#if defined(__has_builtin)
#  if __has_builtin(__builtin_amdgcn_global_load_async_to_lds_b128)
#    define HAVE_ASYNC_LDS 1
#  endif
#endif

#define AS_GLOBAL_V4I(p) ((__attribute__((address_space(1))) v4i*)(p))
#define AS_LOCAL_V4I(p)  ((__attribute__((address_space(3))) v4i*)(p))

// One wave per 128-float row: lane l covers dims [4l, 4l+4).
// Dims 0-31 -> table0, 32-79 -> table1, 80-127 -> table2; the 4-dim chunks
// never straddle tables, so each lane does ONE 16B load per tensor.
// Branch-free: base pointer + offset selected with v_cndmask, single
// divergent-address global_load_b128, nontemporal 16B store.
DEV void emit_pair(const float* __restrict__ c0, const float* __restrict__ c1,
                   const float* __restrict__ c2, const float* __restrict__ s0,
                   const float* __restrict__ s1, const float* __restrict__ s2,
                   int a0, int a1, int a2,
                   float* __restrict__ dstC, float* __restrict__ dstS, int lane) {
  const int off = (lane < 8)  ? a0 * D0 + lane * 4
                : (lane < 20) ? a1 * D1 + (lane - 8) * 4
                              : a2 * D2 + (lane - 20) * 4;
  const float* cb = (lane < 8) ? c0 : (lane < 20) ? c1 : c2;
  const float* sb = (lane < 8) ? s0 : (lane < 20) ? s1 : s2;
  v4f vc = *(const v4f*)(cb + off);
  v4f vs = *(const v4f*)(sb + off);
  __builtin_nontemporal_store(vc, (v4f*)(dstC + lane * 4));
  __builtin_nontemporal_store(vs, (v4f*)(dstS + lane * 4));
}

// ---- img region: [B, 65536, 128]; block = 64 rows, table2 slab async-staged
// into LDS (global_load_async_to_lds_b128 -> s_wait_asynccnt -> ds_load_b128).
__global__ void __launch_bounds__(256)
k_img(const float* __restrict__ c0, const float* __restrict__ c1,
      const float* __restrict__ c2, const float* __restrict__ s0,
      const float* __restrict__ s1, const float* __restrict__ s2,
      const int* __restrict__ clen,
      float* __restrict__ outC, float* __restrict__ outS) {
  __shared__ float l_c2[64 * D2];
  __shared__ float l_s2[64 * D2];
  const int b      = blockIdx.x;                  // 8192 blocks
  const int i      = b >> 10;                     // 1024 blocks / sample
  const int jbase  = (b & 1023) << 6;             // first token of this block
  const int a1     = jbase >> 8;                  // same for all 64 rows
  const int a2base = jbase & 255;                 // a2 = a2base + rr
  const float* g_c2 = c2 + a2base * D2;
  const float* g_s2 = s2 + a2base * D2;
  // Stage 64 rows x 48 floats of table2 (cos & sin): 768 float4 each.
#if HAVE_ASYNC_LDS
  for (int q = threadIdx.x; q < 768; q += 256) {
    __builtin_amdgcn_global_load_async_to_lds_b128(
        AS_GLOBAL_V4I(g_c2 + q * 4), AS_LOCAL_V4I(l_c2 + q * 4), 0, 0);
    __builtin_amdgcn_global_load_async_to_lds_b128(
        AS_GLOBAL_V4I(g_s2 + q * 4), AS_LOCAL_V4I(l_s2 + q * 4), 0, 0);
  }
#  if defined(__has_builtin) && __has_builtin(__builtin_amdgcn_s_wait_asynccnt)
  __builtin_amdgcn_s_wait_asynccnt(0);
#  else
  asm volatile("s_wait_asynccnt 0" ::: "memory");
#  endif
#else
  for (int q = threadIdx.x; q < 768; q += 256) {
    *(v4f*)(l_c2 + q * 4) = *(const v4f*)(g_c2 + q * 4);
    *(v4f*)(l_s2 + q * 4) = *(const v4f*)(g_s2 + q * 4);
  }
#endif
  __syncthreads();
  const int wave = threadIdx.x >> 5, lane = threadIdx.x & 31;
  const int a0 = clen[i] + 256;                   // pe_shift for target image
  for (int rr = wave; rr < 64; rr += 8) {
    const long off = ((long)b * 64 + rr) * D;
    v4f vc, vs;
    if (lane < 20) {                              // global tables 0/1
      const int o = (lane < 8) ? a0 * D0 + lane * 4
                               : a1 * D1 + (lane - 8) * 4;
      const float* cb = (lane < 8) ? c0 : c1;
      const float* sb = (lane < 8) ? s0 : s1;
      vc = *(const v4f*)(cb + o);
      vs = *(const v4f*)(sb + o);
    } else {                                      // table2 from LDS
      const int o = rr * D2 + (lane - 20) * 4;
      vc = *(const v4f*)(l_c2 + o);               // ds_load_b128
      vs = *(const v4f*)(l_s2 + o);
    }
    __builtin_nontemporal_store(vc, (v4f*)(outC + off + lane * 4));
    __builtin_nontemporal_store(vs, (v4f*)(outS + off + lane * 4));
  }
}

// ---- kernel 0: caption lengths (wave32 ballot reduction) -------------------
__global__ void k_caplen(const int* __restrict__ mask, int* __restrict__ clen,
                         float* __restrict__ oCap, float* __restrict__ oSeq) {
  __shared__ int s;
  if (threadIdx.x == 0) s = 0;
  __syncthreads();
  int v = mask[blockIdx.x * ENC + threadIdx.x];
  unsigned long long bal = __ballot(v != 0);
  if ((threadIdx.x & 31) == 0) atomicAdd(&s, (int)__popcll(bal));
  __syncthreads();
  if (threadIdx.x == 0) {
    clen[blockIdx.x] = s;
    oCap[blockIdx.x] = (float)s;
    oSeq[blockIdx.x] = (float)(s + REFT + IMGT);
  }
}

// ---- cap region: [B, 512, 128]; t < c -> (t,t,t), else zeros ---------------
__global__ void __launch_bounds__(256)
k_cap(const float* __restrict__ c0, const float* __restrict__ c1,
      const float* __restrict__ c2, const float* __restrict__ s0,
      const float* __restrict__ s1, const float* __restrict__ s2,
      const int* __restrict__ clen,
      float* __restrict__ outC, float* __restrict__ outS) {
  const int wave = threadIdx.x >> 5, lane = threadIdx.x & 31;
  const int row = blockIdx.x * 8 + wave;          // 0 .. B*ENC-1
  const int i = row >> 9, t = row & (ENC - 1);
  const long off = (long)row * D;
  if (t < clen[i]) {
    emit_pair(c0, c1, c2, s0, s1, s2, t, t, t, outC + off, outS + off, lane);
  } else {
    v4f z = {0.f, 0.f, 0.f, 0.f};
    __builtin_nontemporal_store(z, (v4f*)(outC + off + lane * 4));
    __builtin_nontemporal_store(z, (v4f*)(outS + off + lane * 4));
  }
}

// ---- ref region: [B, 32768, 128] -------------------------------------------
__global__ void __launch_bounds__(256)
k_ref(const float* __restrict__ c0, const float* __restrict__ c1,
      const float* __restrict__ c2, const float* __restrict__ s0,
      const float* __restrict__ s1, const float* __restrict__ s2,
      const int* __restrict__ clen,
      float* __restrict__ outC, float* __restrict__ outS) {
  const int wave = threadIdx.x >> 5, lane = threadIdx.x & 31;
  const int row = blockIdx.x * 8 + wave;          // 0 .. B*REFT-1
  const int i = row >> 15, r = row & (REFT - 1);
  const int c = clen[i];
  const int a0 = c + ((r >> 14) << 7);            // pe_shift = c + 128*k
  const int a1 = (r >> 7) & 127;
  const int a2 = r & 127;
  const long off = (long)row * D;
  emit_pair(c0, c1, c2, s0, s1, s2, a0, a1, a2, outC + off, outS + off, lane);
}

// ---- full freqs: [B, max_seq, 128]; padding rows gather index 0 ------------
__global__ void __launch_bounds__(256)
k_frq(const float* __restrict__ c0, const float* __restrict__ c1,
      const float* __restrict__ c2, const float* __restrict__ s0,
      const float* __restrict__ s1, const float* __restrict__ s2,
      const int* __restrict__ clen,
      float* __restrict__ outC, float* __restrict__ outS,
      int maxseq, int totalRows) {
  const int wave = threadIdx.x >> 5, lane = threadIdx.x & 31;
  const int row = blockIdx.x * 8 + wave;
  if (row >= totalRows) return;
  const int i = row / maxseq;
  const int t = row - i * maxseq;
  const int c = clen[i];
  int a0, a1, a2;
  if (t < c) {
    a0 = a1 = a2 = t;
  } else {
    const int u = t - c;
    if (u < REFT) {                     // reference images
      a0 = c + ((u >> 14) << 7); a1 = (u >> 7) & 127; a2 = u & 127;
    } else if (u < REFT + IMGT) {       // target image
      const int j = u - REFT;
      a0 = c + 256; a1 = j >> 8; a2 = j & 255;
    } else {                            // padding: pos==0 -> cos=1, sin=0
      a0 = a1 = a2 = 0;
    }
  }
  const long off = (long)row * D;
  emit_pair(c0, c1, c2, s0, s1, s2, a0, a1, a2, outC + off, outS + off, lane);
}

extern "C" void kernel_launch(void* const* d_in, const int* in_sizes, int n_in,
                              void* d_out, int out_size, void* d_ws, size_t ws_size,
                              hipStream_t stream) {
  const int*   mask = (const int*)d_in[0];
  const float* c0 = (const float*)d_in[1];
  const float* s0 = (const float*)d_in[2];
  const float* c1 = (const float*)d_in[3];
  const float* s1 = (const float*)d_in[4];
  const float* c2 = (const float*)d_in[5];
  const float* s2 = (const float*)d_in[6];
  float* out  = (float*)d_out;
  int*   clen = (int*)d_ws;

  const long capN = (long)B * ENC  * D;   // one tensor's element count
  const long refN = (long)B * REFT * D;
  const long imgN = (long)B * IMGT * D;
  // out_size = 2*(capN + refN + imgN + B*maxseq*D) + 2*B  -> solve for maxseq
  const long frqN   = ((long)out_size - 2 * B) / 2 - capN - refN - imgN;
  const long maxseq = frqN / ((long)B * D);

  float* capC = out;
  float* capS = capC + capN;
  float* refC = capS + capN;
  float* refS = refC + refN;
  float* imgC = refS + refN;
  float* imgS = imgC + imgN;
  float* frqC = imgS + imgN;
  float* frqS = frqC + frqN;
  float* oCap = frqS + frqN;
  float* oSeq = oCap + B;

  k_caplen<<<B, ENC, 0, stream>>>(mask, clen, oCap, oSeq);
  k_cap<<<(B * ENC) / 8, 256, 0, stream>>>(c0, c1, c2, s0, s1, s2, clen, capC, capS);
  k_ref<<<(B * REFT) / 8, 256, 0, stream>>>(c0, c1, c2, s0, s1, s2, clen, refC, refS);
  k_img<<<(B * IMGT) / 64, 256, 0, stream>>>(c0, c1, c2, s0, s1, s2, clen, imgC, imgS);
  const long frows = (long)B * maxseq;
  k_frq<<<(unsigned)((frows + 7) / 8), 256, 0, stream>>>(
      c0, c1, c2, s0, s1, s2, clen, frqC, frqS, (int)maxseq, (int)frows);
}